// GlobalPointer_22849226014774
// MI455X (gfx1250) — compile-verified
//
#include <hip/hip_runtime.h>
#include <hip/hip_bf16.h>
#include <stdint.h>

// CDNA5 / gfx1250, wave32. fp32 matrix pipe (V_WMMA_F32_16X16X4_F32) +
// async global->LDS staging (ASYNCcnt) + non-temporal output stores.

typedef __attribute__((ext_vector_type(2))) float v2f;
typedef __attribute__((ext_vector_type(8))) float v8f;

#define NUM_HEADS 12
#define HEAD_DIM  64
#define HIDDEN    768
#define BATCH     2
#define SEQ       2048
#define OUTC      (NUM_HEADS * HEAD_DIM * 2)  // 1536
#define ROWS      (BATCH * SEQ)               // 4096
#define NEG_BIG   1000000000000.0f
#define LDS_PITCH 68                          // 64 + 4 pad: bank-conflict-free b64 reads

__device__ __forceinline__ v8f wmma_f32_k4(v2f a, v2f b, v8f c) {
  // 8 args: (neg_a, A, neg_b, B, c_mod, C, reuse_a, reuse_b)
  return __builtin_amdgcn_wmma_f32_16x16x4_f32(false, a, false, b, (short)0, c,
                                               false, false);
}

// Async DMA: 16B global -> LDS, tracked by ASYNCcnt (no VGPR data path).
__device__ __forceinline__ void async_ld_b128(uint32_t lds_off,
                                              const float* gptr) {
  asm volatile("global_load_async_to_lds_b128 %0, %1, off"
               :: "v"(lds_off), "v"(gptr)
               : "memory");
}

__device__ __forceinline__ void wait_async_all() {
  asm volatile("s_wait_asynccnt 0x0" ::: "memory");
}

// ---------------------------------------------------------------------------
// Kernel 1: hs = inputs @ W^T + b   (ROWS x OUTC, K = HIDDEN)
// One wave per 16x16 tile; 192 fp32 WMMAs per tile (K-step 4).
// ---------------------------------------------------------------------------
__global__ void proj_kernel(const float* __restrict__ in,
                            const float* __restrict__ W,
                            const float* __restrict__ bias,
                            float* __restrict__ hs) {
  const int lane = threadIdx.x & 31;
  const int wave = threadIdx.x >> 5;
  const int tile = blockIdx.x * 8 + wave;      // exact: (ROWS/16)*(OUTC/16)
  const int tr = tile / (OUTC / 16);
  const int tc = tile % (OUTC / 16);

  const int mloc = lane & 15;
  const int kp   = (lane >> 4) * 2;
  const float* Arow = in + (size_t)(tr * 16 + mloc) * HIDDEN + kp;
  const float* Brow = W  + (size_t)(tc * 16 + mloc) * HIDDEN + kp;

  v8f acc = {};
#pragma unroll 4
  for (int k = 0; k < HIDDEN; k += 4) {
    v2f a = *(const v2f*)(Arow + k);
    v2f b = *(const v2f*)(Brow + k);
    acc = wmma_f32_k4(a, b, acc);
  }

  const float bv    = bias[tc * 16 + mloc];
  const int   rbase = tr * 16 + (lane >> 4) * 8;
  float* op = hs + (size_t)rbase * OUTC + tc * 16 + mloc;
#pragma unroll
  for (int v = 0; v < 8; ++v) op[(size_t)v * OUTC] = acc[v] + bv;
}

// ---------------------------------------------------------------------------
// Kernel 2: RoPE (reference's half-swap convention) -> q,k in (B,H,N,D)
// ---------------------------------------------------------------------------
__global__ void rope_kernel(const float* __restrict__ hs,
                            float* __restrict__ qb,
                            float* __restrict__ kb) {
  const int idx = blockIdx.x * blockDim.x + threadIdx.x;
  const int i  = idx & 63;
  const int t  = idx >> 6;
  const int h  = t % NUM_HEADS;
  const int bn = t / NUM_HEADS;
  const int n  = bn & (SEQ - 1);
  const int b  = bn >> 11;

  const float* row = hs + (size_t)bn * OUTC + h * (2 * HEAD_DIM);
  const float qv = row[i];
  const float kv = row[HEAD_DIM + i];
  const int   pi  = (i < 32) ? (2 * i + 1) : (2 * (i - 32));
  const float sgn = (i < 32) ? -1.0f : 1.0f;
  const float q2 = sgn * row[pi];
  const float k2 = sgn * row[HEAD_DIM + pi];

  const float theta = __expf(-(float)(i >> 1) * 0.2878231366242557f);
  const float ang = (float)n * theta;
  float s, c;
  __sincosf(ang, &s, &c);

  const size_t o = ((size_t)(b * NUM_HEADS + h) * SEQ + n) * HEAD_DIM + i;
  qb[o] = qv * c + q2 * s;
  kb[o] = kv * c + k2 * s;
}

// ---------------------------------------------------------------------------
// Kernel 3: logits = q @ k^T with masks, *1/8.
// 256-thread block computes a 128(query) x 64(key) macro-tile:
//   - async-DMA q (128x64) and k (64x64) tiles into padded LDS
//   - wave w computes the 16-row strip [w*16, w*16+16) x 4 key sub-tiles
//   - A-fragments held in registers, B-fragments via ds_load_b64
//   - masked result written with non-temporal b32 stores (402MB stream-out)
// ---------------------------------------------------------------------------
__global__ void __launch_bounds__(256)
logits_kernel(const float* __restrict__ qb, const float* __restrict__ kb,
              const int* __restrict__ mask, float* __restrict__ out) {
  __shared__ float qs[128 * LDS_PITCH];
  __shared__ float ks[64 * LDS_PITCH];

  const int lane = threadIdx.x & 31;
  const int wave = threadIdx.x >> 5;

  // grid: 24 * (2048/128) * (2048/64) = 24*16*32 = 12288 blocks
  const int bh = blockIdx.x >> 9;        // / (16*32)
  const int r  = blockIdx.x & 511;
  const int tn = r >> 5;                 // query macro-tile (0..15)
  const int tm = r & 31;                 // key macro-tile   (0..31)

  const float* qg = qb + (size_t)bh * SEQ * HEAD_DIM + (size_t)tn * 128 * HEAD_DIM;
  const float* kg = kb + (size_t)bh * SEQ * HEAD_DIM + (size_t)tm * 64 * HEAD_DIM;

  // ---- async stage q: 128x64 fp32 = 2048 x b128, 8 per thread ----
  const int t = threadIdx.x;
#pragma unroll
  for (int it = 0; it < 8; ++it) {
    const int lin = it * 256 + t;
    const int row = lin >> 4;
    const int c4  = (lin & 15) << 2;
    async_ld_b128((uint32_t)(uintptr_t)&qs[row * LDS_PITCH + c4],
                  qg + row * HEAD_DIM + c4);
  }
  // ---- async stage k: 64x64 fp32 = 1024 x b128, 4 per thread ----
#pragma unroll
  for (int it = 0; it < 4; ++it) {
    const int lin = it * 256 + t;
    const int row = lin >> 4;
    const int c4  = (lin & 15) << 2;
    async_ld_b128((uint32_t)(uintptr_t)&ks[row * LDS_PITCH + c4],
                  kg + row * HEAD_DIM + c4);
  }
  wait_async_all();
  __syncthreads();

  const int mloc = lane & 15;
  const int kp   = (lane >> 4) * 2;

  // A-fragments for this wave's 16-row strip, kept in registers (16 x v2f).
  v2f afrag[16];
#pragma unroll
  for (int s = 0; s < 16; ++s)
    afrag[s] = *(const v2f*)&qs[(wave * 16 + mloc) * LDS_PITCH + 4 * s + kp];

  const int bidx  = bh / NUM_HEADS;
  const int rbase = tn * 128 + wave * 16 + (lane >> 4) * 8;

#pragma unroll
  for (int j = 0; j < 4; ++j) {          // 4 key sub-tiles of 16
    v8f acc = {};
#pragma unroll
    for (int s = 0; s < 16; ++s) {
      v2f b = *(const v2f*)&ks[(j * 16 + mloc) * LDS_PITCH + 4 * s + kp];
      acc = wmma_f32_k4(afrag[s], b, acc);
    }

    const int   col    = tm * 64 + j * 16 + mloc;      // key position m
    const float pad    = (float)mask[bidx * SEQ + col];
    const float padoff = (1.0f - pad) * NEG_BIG;

    float* op = out + ((size_t)bh * SEQ + rbase) * SEQ + col;
#pragma unroll
    for (int v = 0; v < 8; ++v) {
      const int rown = rbase + v;                      // query position n
      float val = acc[v] * pad - padoff;               // key-padding mask
      if (col < rown) val -= NEG_BIG;                  // tril(k=-1) mask
      __builtin_nontemporal_store(val * 0.125f, op + (size_t)v * SEQ);
    }
  }
}

// ---------------------------------------------------------------------------
extern "C" void kernel_launch(void* const* d_in, const int* in_sizes, int n_in,
                              void* d_out, int out_size, void* d_ws,
                              size_t ws_size, hipStream_t stream) {
  const float* inp  = (const float*)d_in[0];   // (B, N, HIDDEN) fp32
  const int*   mask = (const int*)d_in[1];     // (B, N) int32
  const float* W    = (const float*)d_in[2];   // (1536, 768) fp32
  const float* bias = (const float*)d_in[3];   // (1536,) fp32
  float* out = (float*)d_out;                  // (B, H, N, N) fp32

  float* hs = (float*)d_ws;                    // 25MB
  float* qb = hs + (size_t)ROWS * OUTC;        // 12.6MB
  float* kb = qb + (size_t)BATCH * NUM_HEADS * SEQ * HEAD_DIM;  // 12.6MB

  const int proj_blocks = (ROWS / 16) * (OUTC / 16) / 8;                // 3072
  const int rope_blocks = (BATCH * SEQ * NUM_HEADS * HEAD_DIM) / 256;   // 12288
  const int logit_blocks =
      BATCH * NUM_HEADS * (SEQ / 128) * (SEQ / 64);                     // 12288

  proj_kernel<<<proj_blocks, 256, 0, stream>>>(inp, W, bias, hs);
  rope_kernel<<<rope_blocks, 256, 0, stream>>>(hs, qb, kb);
  logits_kernel<<<logit_blocks, 256, 0, stream>>>(qb, kb, mask, out);
}